// Decoder_61959198212561
// MI455X (gfx1250) — compile-verified
//
#include <hip/hip_runtime.h>

#define NEG_SLOPE 0.2f

typedef __attribute__((ext_vector_type(2))) float v2f;
typedef __attribute__((ext_vector_type(8))) float v8f;

// ---------- helpers: order-preserving float<->uint mapping for atomicMax ----
__device__ __forceinline__ unsigned f2sortable(float x) {
    unsigned b = __float_as_uint(x);
    return (b & 0x80000000u) ? ~b : (b | 0x80000000u);
}
__device__ __forceinline__ float sortable2f(unsigned s) {
    unsigned b = (s & 0x80000000u) ? (s & 0x7FFFFFFFu) : ~s;
    return __uint_as_float(b);
}
#define SORTABLE_NEG_INF 0x007FFFFFu   // f2sortable(-inf)

// ---------- WMMA fp32 GEMM: C[M,Nout] = A[M,K] @ W[Nout,K]^T (+bias, relu) --
// One wave computes one 16x16 tile via V_WMMA_F32_16X16X4_F32.
// Branch-free hot loop: A rows valid by clamp (M%16==0 in practice), B row
// pointer clamped (OOB columns discarded at store), only the K-tail (K%4!=0,
// layer 1 and final fc) needs a value-select on A.
__global__ void gemm_wmma_kernel(const float* __restrict__ A,
                                 const float* __restrict__ W,    // [Nout,K]
                                 const float* __restrict__ bias, // or nullptr
                                 float* __restrict__ C,
                                 int M, int K, int Nout, int relu)
{
    const int lane  = threadIdx.x;            // 0..31
    const int mtile = blockIdx.y * blockDim.y + threadIdx.y;
    const int m0 = mtile * 16;
    const int n0 = blockIdx.x * 16;

    const int r  = lane & 15;
    const int hi = lane >> 4;                 // 0: K+0/K+1, 1: K+2/K+3
    const int kb = hi * 2;

    const int arow = min(m0 + r, M - 1);      // clamp (full tiles when M%16==0)
    const int bcol = min(n0 + r, Nout - 1);   // clamp OOB columns
    const float* __restrict__ Arow = A + (size_t)arow * K;
    const float* __restrict__ Brow = W + (size_t)bcol * K;

    v8f acc = {0.f, 0.f, 0.f, 0.f, 0.f, 0.f, 0.f, 0.f};

    const int kfull = K & ~3;
    for (int k = 0; k < kfull; k += 4) {
        v2f a, b;
        a.x = Arow[k + kb];
        a.y = Arow[k + kb + 1];
        b.x = Brow[k + kb];
        b.y = Brow[k + kb + 1];
        acc = __builtin_amdgcn_wmma_f32_16x16x4_f32(
            false, a, false, b, (short)0, acc, false, false);
    }
    if (kfull < K) {                          // uniform tail (K % 4 != 0)
        const int k0 = kfull + kb;
        const int k1 = k0 + 1;
        const float ax = Arow[min(k0, K - 1)];   // unconditional clamped loads
        const float ay = Arow[min(k1, K - 1)];
        v2f a, b;
        a.x = (k0 < K) ? ax : 0.f;               // v_cndmask, no branch
        a.y = (k1 < K) ? ay : 0.f;
        b.x = Brow[min(k0, K - 1)];              // garbage ok: A-tail is zero
        b.y = Brow[min(k1, K - 1)];
        acc = __builtin_amdgcn_wmma_f32_16x16x4_f32(
            false, a, false, b, (short)0, acc, false, false);
    }

    const int cn = n0 + r;
    if (cn < Nout) {
        const float bv = bias ? bias[cn] : 0.f;
        const int mb = m0 + hi * 8;
        #pragma unroll
        for (int i = 0; i < 8; ++i) {
            const int cm = mb + i;
            if (cm < M) {
                float v = acc[i] + bv;
                if (relu) v = fmaxf(v, 0.f);
                C[(size_t)cm * Nout + cn] = v;
            }
        }
    }
}

// ---------- attention dot products: el/er [N,H] from h [N,H,D] --------------
template <int H, int D>
__global__ void attn_dots_kernel(const float* __restrict__ h,
                                 const float* __restrict__ al,
                                 const float* __restrict__ ar,
                                 float* __restrict__ el,
                                 float* __restrict__ er,
                                 int N)
{
    const int idx = blockIdx.x * blockDim.x + threadIdx.x;
    if (idx >= N * H) return;
    const int n = idx / H, hh = idx % H;
    const float* hp  = h + (size_t)n * (H * D) + hh * D;
    const float* alp = al + hh * D;
    const float* arp = ar + hh * D;
    float sl = 0.f, sr = 0.f;
    #pragma unroll
    for (int d = 0; d < D; ++d) {
        const float v = hp[d];
        sl += v * alp[d];
        sr += v * arp[d];
    }
    el[idx] = sl;
    er[idx] = sr;
}

// ---------- fills -----------------------------------------------------------
__global__ void fill_u32_kernel(unsigned* __restrict__ p, unsigned v, int n) {
    const int i = blockIdx.x * blockDim.x + threadIdx.x;
    if (i < n) p[i] = v;
}
__global__ void fill_f32_kernel(float* __restrict__ p, float v, int n) {
    const int i = blockIdx.x * blockDim.x + threadIdx.x;
    if (i < n) p[i] = v;
}

// ---------- edge pass 1: e = lrelu(el[src]+er[dst]); segment max ------------
template <int H>
__global__ void edge_max_kernel(const int* __restrict__ src,
                                const int* __restrict__ dst,
                                const float* __restrict__ el,
                                const float* __restrict__ er,
                                float* __restrict__ ebuf,
                                unsigned* __restrict__ emax,
                                int E)
{
    const int idx = blockIdx.x * blockDim.x + threadIdx.x;
    if (idx >= E * H) return;
    const int e = idx / H, hh = idx % H;
    const int s = src[e], d = dst[e];
    float x = el[s * H + hh] + er[d * H + hh];
    x = (x > 0.f) ? x : NEG_SLOPE * x;
    ebuf[idx] = x;
    atomicMax(&emax[d * H + hh], f2sortable(x));
}

// ---------- decode sortable max, replace -inf (no incoming edges) with 0 ----
__global__ void emax_fix_kernel(float* __restrict__ emax, int n)
{
    const int i = blockIdx.x * blockDim.x + threadIdx.x;
    if (i >= n) return;
    const unsigned s = __float_as_uint(emax[i]);
    emax[i] = (s == SORTABLE_NEG_INF) ? 0.f : sortable2f(s);
}

// ---------- edge pass 2: a = exp(e - emax[dst]); segment sum ----------------
template <int H>
__global__ void edge_exp_kernel(const int* __restrict__ dst,
                                const float* __restrict__ emax,
                                float* __restrict__ ebuf,   // in: e, out: a
                                float* __restrict__ denom,
                                int E)
{
    const int idx = blockIdx.x * blockDim.x + threadIdx.x;
    if (idx >= E * H) return;
    const int e = idx / H, hh = idx % H;
    const int d = dst[e];
    const float a = expf(ebuf[idx] - emax[d * H + hh]);
    ebuf[idx] = a;
    atomicAdd(&denom[d * H + hh], a);
}

// ---------- edge pass 3: out[dst] += (a/denom[dst]) * h[src] ----------------
template <int H, int D>
__global__ void edge_scatter_kernel(const int* __restrict__ src,
                                    const int* __restrict__ dst,
                                    const float* __restrict__ ebuf,   // a
                                    const float* __restrict__ denom,
                                    const float* __restrict__ h,      // [N,H,D]
                                    float* __restrict__ out,          // [N,H,D]
                                    int E)
{
    const int idx = blockIdx.x * blockDim.x + threadIdx.x;
    if (idx >= E * H) return;
    const int e = idx / H, hh = idx % H;
    const int s = src[e], d = dst[e];
    const float alpha = ebuf[idx] / denom[d * H + hh];
    const float* hs = h   + (size_t)s * (H * D) + hh * D;
    float*       od = out + (size_t)d * (H * D) + hh * D;
    #pragma unroll
    for (int dd = 0; dd < D; ++dd)
        atomicAdd(od + dd, alpha * hs[dd]);
}

// ---------- bias + relu -----------------------------------------------------
template <int F>
__global__ void bias_relu_kernel(const float* __restrict__ acc,
                                 const float* __restrict__ bias,
                                 float* __restrict__ out,
                                 int N)
{
    const int i = blockIdx.x * blockDim.x + threadIdx.x;
    if (i >= N * F) return;
    out[i] = fmaxf(acc[i] + bias[i % F], 0.f);
}

// ---------- host-side orchestration -----------------------------------------
static inline int ceil_div(int a, int b) { return (a + b - 1) / b; }

static void launch_gemm(const float* A, const float* W, const float* bias,
                        float* C, int M, int K, int Nout, int relu,
                        hipStream_t stream)
{
    dim3 block(32, 4);
    dim3 grid(ceil_div(Nout, 16), ceil_div(M, 16 * 4));
    gemm_wmma_kernel<<<grid, block, 0, stream>>>(A, W, bias, C, M, K, Nout, relu);
}

template <int H, int D>
static void run_gat_layer(const float* x, int Fin,
                          const float* W, const float* al, const float* ar,
                          const float* bias,
                          const int* src, const int* dst, int N, int E,
                          float* hbuf, float* accb, float* xout,
                          float* el, float* er, float* emax, float* denom,
                          float* ebuf, hipStream_t stream)
{
    constexpr int HD = H * D;
    const int NH  = N * H;
    const int EH  = E * H;
    const int NHD = N * HD;

    // h = x @ W^T
    launch_gemm(x, W, nullptr, hbuf, N, Fin, HD, 0, stream);
    // el, er
    attn_dots_kernel<H, D><<<ceil_div(NH, 256), 256, 0, stream>>>(hbuf, al, ar, el, er, N);
    // init segment buffers
    fill_u32_kernel<<<ceil_div(NH, 256), 256, 0, stream>>>((unsigned*)emax, SORTABLE_NEG_INF, NH);
    fill_f32_kernel<<<ceil_div(NH, 256), 256, 0, stream>>>(denom, 0.f, NH);
    fill_f32_kernel<<<ceil_div(NHD, 256), 256, 0, stream>>>(accb, 0.f, NHD);
    // edge softmax + aggregation
    edge_max_kernel<H><<<ceil_div(EH, 256), 256, 0, stream>>>(src, dst, el, er, ebuf, (unsigned*)emax, E);
    emax_fix_kernel<<<ceil_div(NH, 256), 256, 0, stream>>>(emax, NH);
    edge_exp_kernel<H><<<ceil_div(EH, 256), 256, 0, stream>>>(dst, emax, ebuf, denom, E);
    edge_scatter_kernel<H, D><<<ceil_div(EH, 256), 256, 0, stream>>>(src, dst, ebuf, denom, hbuf, accb, E);
    // out = relu(acc + bias)
    bias_relu_kernel<HD><<<ceil_div(NHD, 256), 256, 0, stream>>>(accb, bias, xout, N);
}

extern "C" void kernel_launch(void* const* d_in, const int* in_sizes, int n_in,
                              void* d_out, int out_size, void* d_ws, size_t ws_size,
                              hipStream_t stream)
{
    const float* features = (const float*)d_in[0];   // [N,25]
    const int*   src      = (const int*)  d_in[1];   // [E]
    const int*   dst      = (const int*)  d_in[2];   // [E]
    const float* W1  = (const float*)d_in[3];
    const float* al1 = (const float*)d_in[4];
    const float* ar1 = (const float*)d_in[5];
    const float* b1  = (const float*)d_in[6];
    const float* W2  = (const float*)d_in[7];
    const float* al2 = (const float*)d_in[8];
    const float* ar2 = (const float*)d_in[9];
    const float* b2  = (const float*)d_in[10];
    const float* W3  = (const float*)d_in[11];
    const float* al3 = (const float*)d_in[12];
    const float* ar3 = (const float*)d_in[13];
    const float* b3  = (const float*)d_in[14];
    const float* fcw = (const float*)d_in[15];
    const float* fcb = (const float*)d_in[16];

    const int N = in_sizes[0] / 25;
    const int E = in_sizes[1];

    // workspace layout (fp32)
    float* ws = (float*)d_ws;
    const size_t N100 = (size_t)N * 100;
    float* nbuf  = ws;                 // layer input / result      [N,100]
    float* hbuf  = nbuf  + N100;       // post-GEMM h               [N,100]
    float* accb  = hbuf  + N100;       // aggregation accumulator   [N,100]
    float* el    = accb  + N100;       // [N,4]
    float* er    = el    + (size_t)N * 4;
    float* emax  = er    + (size_t)N * 4;
    float* denom = emax  + (size_t)N * 4;
    float* ebuf  = denom + (size_t)N * 4;   // [E,4]

    // layer 1: 25 -> (4 heads x 10)
    run_gat_layer<4, 10>(features, 25, W1, al1, ar1, b1,
                         src, dst, N, E, hbuf, accb, nbuf,
                         el, er, emax, denom, ebuf, stream);
    // layer 2: 40 -> (4 heads x 25)
    run_gat_layer<4, 25>(nbuf, 40, W2, al2, ar2, b2,
                         src, dst, N, E, hbuf, accb, nbuf,
                         el, er, emax, denom, ebuf, stream);
    // layer 3: 100 -> (1 head x 50)
    run_gat_layer<1, 50>(nbuf, 100, W3, al3, ar3, b3,
                         src, dst, N, E, hbuf, accb, nbuf,
                         el, er, emax, denom, ebuf, stream);
    // final fc: 50 -> 93 (bias, no relu)
    launch_gemm(nbuf, fcw, fcb, (float*)d_out, N, 50, 93, 0, stream);
}